// GELU79_64458869179064
// MI455X (gfx1250) — compile-verified
//
#include <hip/hip_runtime.h>
#include <math.h>

// MI455X / gfx1250, wave32. Memory-bound (~402 MB HBM traffic min -> ~17us at
// 23.3 TB/s). fp32 WMMA (V_WMMA_F32_16X16X4_F32) computes the 8192x4096 GEMV
// (cos-sim dot) exactly in fp32; row stats fused into the same x-stream.
// gelu via sigmoid form with v_exp_f32 + v_rcp_f32: branch-free, EXEC uniform.

typedef float v2f __attribute__((ext_vector_type(2)));
typedef float v8f __attribute__((ext_vector_type(8)));

#define D_DIM 4096
#define T_DIM 2048

// Branch-free tanh: 1 - 2*rcp(e^{2x}+1). Saturates to +/-1 for |x| -> inf.
__device__ __forceinline__ float fast_tanh(float x) {
    return 1.0f - 2.0f * __builtin_amdgcn_rcpf(__expf(2.0f * x) + 1.0f);
}

// gelu(v) = 0.5 v (1 + tanh(z)), z = c0*(v + c1 v^3)
//         = v * sigmoid(2z) = v * (1 - rcp(e^{2z} + 1))
__device__ __forceinline__ float gelu_f(float v) {
    float u  = v * v;
    float tz = v * fmaf(0.07135481627f, u, 1.5957691216057308f);   // 2*z
    float r  = __builtin_amdgcn_rcpf(__expf(tz) + 1.0f);
    return v - v * r;
}

__device__ __forceinline__ float wave_reduce_add(float v) {
#pragma unroll
    for (int m = 16; m >= 1; m >>= 1) v += __shfl_xor(v, m, 32);
    return v;
}

// ---------------------------------------------------------------------------
// K0: norms of the three D-length state vectors + scalar transforms.
// ---------------------------------------------------------------------------
__global__ void k0_scalars(const float* __restrict__ ema_out,
                           const float* __restrict__ ema_x,
                           const float* __restrict__ ema_raw,
                           const float* __restrict__ p_ld_l,
                           const float* __restrict__ p_log_tau,
                           const float* __restrict__ p_lsv,
                           const float* __restrict__ p_lsg,
                           const float* __restrict__ p_lwv,
                           const float* __restrict__ p_lwg,
                           float* __restrict__ ws_scal) {
    __shared__ float sh[3][8];
    int tid = threadIdx.x;
    float s0 = 0.f, s1 = 0.f, s2 = 0.f;
    for (int i = tid; i < D_DIM; i += 256) {
        float a = ema_out[i]; s0 += a * a;
        float b = ema_x[i];   s1 += b * b;
        float c = ema_raw[i]; s2 += c * c;
    }
    s0 = wave_reduce_add(s0);
    s1 = wave_reduce_add(s1);
    s2 = wave_reduce_add(s2);
    int wid = tid >> 5, lane = tid & 31;
    if (lane == 0) { sh[0][wid] = s0; sh[1][wid] = s1; sh[2][wid] = s2; }
    __syncthreads();
    if (tid == 0) {
        float t0 = 0.f, t1 = 0.f, t2 = 0.f;
#pragma unroll
        for (int i = 0; i < 8; i++) { t0 += sh[0][i]; t1 += sh[1][i]; t2 += sh[2][i]; }
        float n_eo = sqrtf(t0), n_ex = sqrtf(t1), n_raw = sqrtf(t2);
        ws_scal[0] = 1.0f / (1.0f + expf(-p_ld_l[0]));   // d_l = sigmoid(logit_decay_l)
        ws_scal[1] = expf(p_log_tau[0]);                 // tau
        ws_scal[2] = log1pf(expf(p_lsv[0]));             // sig_v = softplus
        ws_scal[3] = log1pf(expf(p_lsg[0]));             // sig_g
        ws_scal[4] = log1pf(expf(p_lwv[0]));             // w_v
        ws_scal[5] = log1pf(expf(p_lwg[0]));             // w_g
        ws_scal[6] = 1.0f / fmaxf(n_eo, 1e-12f);         // 1/max(||ema_out||,1e-12)
        ws_scal[7] = n_ex + 2e-5f;                       // (||ema_x||+EPS)+EPS
        ws_scal[8] = n_raw + 2e-5f;                      // (||ema_raw||+EPS)+EPS
    }
}

// ---------------------------------------------------------------------------
// K1: single pass over x producing, per row:
//   ws_dot  = dot(gelu(x_row), ema_out)   via V_WMMA_F32_16X16X4_F32
//   ws_out2 = ||gelu(x_row)||^2           fused into the same stream
//   ws_dx2  = ||x_row - ema_x||^2
// A-tile (16x4 f32): lane L<16 -> row L, K={0,1}; lane L+16 -> K={2,3};
// i.e. a per-lane float2 at kb + 2*(lane>>4). B broadcasts the ema vector to
// all 16 columns -> identical per-lane addressing, served from LDS.
// Lane pair (L, L+16) covers complementary K-halves, so the scalar partials
// combine with one shfl_xor(.,16). Column N=0 of D holds the 16 row-dots.
// ---------------------------------------------------------------------------
__global__ void __launch_bounds__(128)
k1_wmma_stats(const float* __restrict__ x,
              const float* __restrict__ ema_out,
              const float* __restrict__ ema_x,
              float* __restrict__ ws_dot,
              float* __restrict__ ws_out2,
              float* __restrict__ ws_dx2) {
    __shared__ float se[D_DIM];   // ema_out
    __shared__ float sx[D_DIM];   // ema_x
    int tid = threadIdx.x;
    for (int i = tid * 4; i < D_DIM; i += 128 * 4) {
        *reinterpret_cast<float4*>(&se[i]) = *reinterpret_cast<const float4*>(&ema_out[i]);
        *reinterpret_cast<float4*>(&sx[i]) = *reinterpret_cast<const float4*>(&ema_x[i]);
    }
    __syncthreads();

    int wid = tid >> 5, lane = tid & 31;
    int base_row = (blockIdx.x * 4 + wid) * 16;
    const float* xr = x + (size_t)(base_row + (lane & 15)) * D_DIM;
    int koff = (lane >> 4) << 1;

    v8f c0 = {}; v8f c1 = {};
    float o2 = 0.f, dx2 = 0.f;
    for (int kb = 0; kb < D_DIM; kb += 8) {
        float2 xa0 = *reinterpret_cast<const float2*>(xr + kb + koff);
        float2 xa1 = *reinterpret_cast<const float2*>(xr + kb + 4 + koff);
        float2 eb0 = *reinterpret_cast<const float2*>(&se[kb + koff]);
        float2 eb1 = *reinterpret_cast<const float2*>(&se[kb + 4 + koff]);
        float2 ex0 = *reinterpret_cast<const float2*>(&sx[kb + koff]);
        float2 ex1 = *reinterpret_cast<const float2*>(&sx[kb + 4 + koff]);
        float g00 = gelu_f(xa0.x), g01 = gelu_f(xa0.y);
        float g10 = gelu_f(xa1.x), g11 = gelu_f(xa1.y);
        o2 += g00 * g00 + g01 * g01 + g10 * g10 + g11 * g11;
        float d;
        d = xa0.x - ex0.x; dx2 += d * d;
        d = xa0.y - ex0.y; dx2 += d * d;
        d = xa1.x - ex1.x; dx2 += d * d;
        d = xa1.y - ex1.y; dx2 += d * d;
        v2f a0 = { g00, g01 };
        v2f b0 = { eb0.x, eb0.y };
        v2f a1 = { g10, g11 };
        v2f b1 = { eb1.x, eb1.y };
        c0 = __builtin_amdgcn_wmma_f32_16x16x4_f32(false, a0, false, b0, (short)0, c0, false, false);
        c1 = __builtin_amdgcn_wmma_f32_16x16x4_f32(false, a1, false, b1, (short)0, c1, false, false);
    }
    // combine complementary K-halves held by lanes L and L+16
    o2  += __shfl_xor(o2, 16, 32);
    dx2 += __shfl_xor(dx2, 16, 32);
    v8f cs = c0 + c1;
    if (lane < 16) {
        ws_out2[base_row + lane] = o2;
        ws_dx2[base_row + lane]  = dx2;
    }
    if (lane == 0) {
#pragma unroll
        for (int j = 0; j < 8; j++) ws_dot[base_row + j] = cs[j];        // rows 0..7
    } else if (lane == 16) {
#pragma unroll
        for (int j = 0; j < 8; j++) ws_dot[base_row + 8 + j] = cs[j];    // rows 8..15
    }
}

// ---------------------------------------------------------------------------
// K2: sequential EMA scan over T (parallel over B x D), fused with the final
// gate multiply so `out` hits HBM exactly once. 1024 threads = 32 waves hold
// the 4096-wide EMA state in registers (float4/thread). The t+1 load is issued
// before the t reduction/barriers so global latency hides behind the sync.
// ---------------------------------------------------------------------------
__global__ void __launch_bounds__(1024)
k2_scan_gate(const float* __restrict__ x,
             const float* __restrict__ ema_raw,
             const float* __restrict__ ws_scal,
             const float* __restrict__ ws_dot,
             const float* __restrict__ ws_out2,
             const float* __restrict__ ws_dx2,
             float* __restrict__ out) {
    __shared__ float sred[32];
    __shared__ float sgate;
    int b = blockIdx.x;
    int tid = threadIdx.x;
    int wid = tid >> 5, lane = tid & 31;
    int col = tid * 4;

    float d_l = ws_scal[0], tau = ws_scal[1], sig_v = ws_scal[2], sig_g = ws_scal[3];
    float w_v = ws_scal[4], w_g = ws_scal[5], inv_ne = ws_scal[6];
    float den_g = ws_scal[7], den_v = ws_scal[8];
    float one_m = 1.0f - d_l;

    float4 ema = *reinterpret_cast<const float4*>(&ema_raw[col]);
    const float* xb = x + (size_t)b * T_DIM * D_DIM;
    float* ob = out + (size_t)b * T_DIM * D_DIM;

    float4 xv = *reinterpret_cast<const float4*>(xb + col);   // t = 0
    for (int t = 0; t < T_DIM; ++t) {
        float4 o;
        o.x = gelu_f(xv.x); o.y = gelu_f(xv.y); o.z = gelu_f(xv.z); o.w = gelu_f(xv.w);
        // prefetch next timestep before the block reduction hides us in barriers
        int tn = (t + 1 < T_DIM) ? (t + 1) : t;
        float4 xnext = *reinterpret_cast<const float4*>(xb + (size_t)tn * D_DIM + col);
        float d, p = 0.f;
        d = o.x - ema.x; p += d * d;
        d = o.y - ema.y; p += d * d;
        d = o.z - ema.z; p += d * d;
        d = o.w - ema.w; p += d * d;
        p = wave_reduce_add(p);
        if (lane == 0) sred[wid] = p;
        __syncthreads();
        if (tid == 0) {
            float tot = 0.f;
#pragma unroll
            for (int i = 0; i < 32; i++) tot += sred[i];
            int row = b * T_DIM + t;
            float no = fmaxf(sqrtf(ws_out2[row]), 1e-12f);
            float cosv = fminf(fmaxf(ws_dot[row] * inv_ne / no, -1.0f), 1.0f);
            float gc  = __expf(-tau * cosv);
            float sg  = fast_tanh(sig_g * sqrtf(ws_dx2[row]) / den_g);
            float vel = fast_tanh(sig_v * sqrtf(tot) / den_v);
            sgate = gc * (1.0f + w_v * vel + w_g * sg);
        }
        __syncthreads();
        float g = sgate;
        float4 r;
        r.x = o.x * g; r.y = o.y * g; r.z = o.z * g; r.w = o.w * g;
        *reinterpret_cast<float4*>(ob + (size_t)t * D_DIM + col) = r;
        ema.x = d_l * ema.x + one_m * o.x;
        ema.y = d_l * ema.y + one_m * o.y;
        ema.z = d_l * ema.z + one_m * o.z;
        ema.w = d_l * ema.w + one_m * o.w;
        xv = xnext;
    }
}

// ---------------------------------------------------------------------------
extern "C" void kernel_launch(void* const* d_in, const int* in_sizes, int n_in,
                              void* d_out, int out_size, void* d_ws, size_t ws_size,
                              hipStream_t stream) {
    const float* x       = (const float*)d_in[0];
    const float* ema_out = (const float*)d_in[1];
    const float* ema_x   = (const float*)d_in[2];
    const float* ema_raw = (const float*)d_in[3];
    // d_in[4] = logit_decay_g (unused by the reference output path)
    const float* ld_l    = (const float*)d_in[5];
    const float* log_tau = (const float*)d_in[6];
    const float* lsv     = (const float*)d_in[7];
    const float* lsg     = (const float*)d_in[8];
    const float* lwv     = (const float*)d_in[9];
    const float* lwg     = (const float*)d_in[10];
    float* out = (float*)d_out;
    float* ws  = (float*)d_ws;

    int BT = in_sizes[0] / D_DIM;   // B*T = 8192
    int B  = BT / T_DIM;            // 4

    float* ws_scal = ws;            // 16 floats
    float* ws_dot  = ws + 16;       // BT floats
    float* ws_out2 = ws + 16 + BT;  // BT floats
    float* ws_dx2  = ws + 16 + 2 * BT;

    k0_scalars<<<1, 256, 0, stream>>>(ema_out, ema_x, ema_raw, ld_l, log_tau,
                                      lsv, lsg, lwv, lwg, ws_scal);
    k1_wmma_stats<<<BT / 64, 128, 0, stream>>>(x, ema_out, ema_x,
                                               ws_dot, ws_out2, ws_dx2);
    k2_scan_gate<<<B, 1024, 0, stream>>>(x, ema_raw, ws_scal, ws_dot,
                                         ws_out2, ws_dx2, out);
}